// RNNAttn_32160715112782
// MI455X (gfx1250) — compile-verified
//
#include <hip/hip_runtime.h>
#include <math.h>

#define Vv 32000
#define Ee 32
#define Hh 128
#define Bb 8
#define Tt 256

typedef float v2f __attribute__((ext_vector_type(2)));
typedef float v8f __attribute__((ext_vector_type(8)));

// ---------------------------------------------------------------------------
// Kernel 1: recurrent attention RNN. One block per batch element (fully
// independent recurrences). keys are computed incrementally (once per step)
// instead of recomputing hid_buf @ U_attn every step (8.6 GMAC -> 33.5 MMAC).
// ---------------------------------------------------------------------------
__global__ __launch_bounds__(256)
void rnn_attn_recurrent(const int* __restrict__ x,
                        const float* __restrict__ emb,
                        const float* __restrict__ W_attn,
                        const float* __restrict__ U_attn,
                        const float* __restrict__ v_attn,
                        const float* __restrict__ W_ih,
                        const float* __restrict__ b_ih,
                        const float* __restrict__ W_hh,
                        const float* __restrict__ b_hh,
                        float* __restrict__ hid,     // [B*T, H]  (also GEMM A)
                        float* __restrict__ keys,    // [B*T, H]
                        float* __restrict__ h_final) // [B, H]
{
    const int b   = blockIdx.x;
    const int tid = threadIdx.x;

    __shared__ float h_cur[Hh];
    __shared__ float q[Hh];
    __shared__ float base[Hh];
    __shared__ float ctx[Hh];
    __shared__ float sc[Tt];
    __shared__ float red[256];

    if (tid < Hh) h_cur[tid] = 0.0f;
    __syncthreads();

    for (int t = 0; t < Tt; ++t) {
        // ---- Phase A: q = h @ W_attn (threads 0..127)
        //               base = b_ih + b_hh + h@W_hh + e_t@W_ih_top (threads 128..255)
        if (tid < Hh) {
            float acc = 0.f;
            #pragma unroll 4
            for (int k = 0; k < Hh; ++k) acc += h_cur[k] * W_attn[k * Hh + tid];
            q[tid] = acc;
        } else {
            const int j = tid - Hh;
            float acc = b_ih[j] + b_hh[j];
            #pragma unroll 4
            for (int k = 0; k < Hh; ++k) acc += h_cur[k] * W_hh[k * Hh + j];
            const int tok = x[b * Tt + t];
            #pragma unroll 4
            for (int k = 0; k < Ee; ++k) acc += emb[tok * Ee + k] * W_ih[k * Hh + j];
            base[j] = acc;
        }
        __syncthreads();

        if (t > 0) {
            // ---- Phase B: scores[t'] = v . tanh(q + keys[t'])  for t' < t
            for (int tp = tid; tp < t; tp += 256) {
                const float* kr = keys + (size_t)(b * Tt + tp) * Hh;
                float s = 0.f;
                #pragma unroll 4
                for (int k = 0; k < Hh; ++k) s += v_attn[k] * tanhf(q[k] + kr[k]);
                sc[tp] = s;
            }
            __syncthreads();

            // ---- Phase C: softmax over sc[0..t)
            float pm = -1e30f;
            for (int tp = tid; tp < t; tp += 256) pm = fmaxf(pm, sc[tp]);
            red[tid] = pm;
            __syncthreads();
            for (int off = 128; off > 0; off >>= 1) {
                if (tid < off) red[tid] = fmaxf(red[tid], red[tid + off]);
                __syncthreads();
            }
            const float m = red[0];
            __syncthreads();
            float ps = 0.f;
            for (int tp = tid; tp < t; tp += 256) {
                const float e = __expf(sc[tp] - m);
                sc[tp] = e;
                ps += e;
            }
            red[tid] = ps;
            __syncthreads();
            for (int off = 128; off > 0; off >>= 1) {
                if (tid < off) red[tid] += red[tid + off];
                __syncthreads();
            }
            const float inv = 1.0f / red[0];
            __syncthreads();

            // ---- Phase D: ctx = sum_t' w[t'] * hid[t']
            if (tid < Hh) {
                const float* hb = hid + (size_t)(b * Tt) * Hh + tid;
                float c = 0.f;
                for (int tp = 0; tp < t; ++tp) c += sc[tp] * hb[(size_t)tp * Hh];
                ctx[tid] = c * inv;
            }
        } else {
            if (tid < Hh) ctx[tid] = 0.f;  // t == 0: zero context
        }
        __syncthreads();

        // ---- Phase E: h_new = tanh(base + ctx @ W_ih_bottom)
        if (tid < Hh) {
            float v = base[tid];
            #pragma unroll 4
            for (int k = 0; k < Hh; ++k) v += ctx[k] * W_ih[(Ee + k) * Hh + tid];
            const float hn = tanhf(v);
            h_cur[tid] = hn;
            hid[(size_t)(b * Tt + t) * Hh + tid] = hn;
        }
        __syncthreads();

        // ---- Phase F: keys[t] = h_new @ U_attn (incremental, computed once)
        if (tid < Hh) {
            float acc = 0.f;
            #pragma unroll 4
            for (int k = 0; k < Hh; ++k) acc += h_cur[k] * U_attn[k * Hh + tid];
            keys[(size_t)(b * Tt + t) * Hh + tid] = acc;
        }
        __syncthreads();
    }

    if (tid < Hh) h_final[b * Hh + tid] = h_cur[tid];
}

// ---------------------------------------------------------------------------
// Kernel 2: logits = acts[2048,128] @ W_out[128,32000] + b_out
// f32 WMMA (V_WMMA_F32_16X16X4_F32): kernel is store-bandwidth-bound
// (262 MB of logits ~ 11us @ 23.3 TB/s), so full-f32 matrix ops cost nothing
// versus low precision and match the reference numerics.
// Each wave: 16(M) x 64(N) tile, 4 v8f accumulators, K=128 in steps of 4.
// Logits stores are NON-TEMPORAL: 262 MB output > 192 MB L2, so streaming
// it past L2 keeps W_out (16 MB, reused by all 128 M-tile rows) resident.
// ---------------------------------------------------------------------------
__global__ __launch_bounds__(128)
void logits_gemm_wmma(const float* __restrict__ A,    // [2048, 128]
                      const float* __restrict__ W,    // [128, 32000]
                      const float* __restrict__ bias, // [32000]
                      float* __restrict__ C)          // [2048, 32000]
{
    constexpr int N = Vv, K = Hh;
    const int lane = threadIdx.x & 31;
    const int wave = threadIdx.x >> 5;
    const int half = lane >> 4;     // 0: K pair {0,1}, 1: K pair {2,3}
    const int l16  = lane & 15;     // M row (A) / N col (B,C,D)
    const int m0 = blockIdx.y * 16;
    const int n0 = blockIdx.x * 256 + wave * 64;

    // Fold bias into the accumulator: every C element in this lane's register
    // file shares the lane's column, so broadcast bias across all 8 rows.
    v8f acc[4];
    #pragma unroll
    for (int s = 0; s < 4; ++s) {
        const float bb = bias[n0 + s * 16 + l16];
        #pragma unroll
        for (int r = 0; r < 8; ++r) acc[s][r] = bb;
    }

    // A fragment: VGPR0 = A[m][k + 2*half], VGPR1 = A[m][k + 2*half + 1]
    const float* Ap = A + (size_t)(m0 + l16) * K + 2 * half;
    // B fragment: VGPR0 = W[k + 2*half][n], VGPR1 = W[k + 2*half + 1][n]
    const float* Bp = W + (size_t)(2 * half) * N + n0 + l16;

    #pragma unroll 8
    for (int k = 0; k < K; k += 4) {
        v2f a;
        a.x = Ap[k];
        a.y = Ap[k + 1];
        const float* B0 = Bp + (size_t)k * N;
        #pragma unroll
        for (int s = 0; s < 4; ++s) {
            v2f bv;
            bv.x = B0[s * 16];
            bv.y = B0[s * 16 + N];
            acc[s] = __builtin_amdgcn_wmma_f32_16x16x4_f32(
                /*neg_a=*/false, a, /*neg_b=*/false, bv,
                /*c_mod=*/(short)0, acc[s],
                /*reuse_a=*/false, /*reuse_b=*/false);
        }
    }

    // C/D layout: VGPR r -> row m0 + half*8 + r, col n0 + s*16 + l16.
    // Non-temporal: write-once streaming output, do not pollute L2.
    float* Cp = C + (size_t)(m0 + half * 8) * N + n0 + l16;
    #pragma unroll
    for (int s = 0; s < 4; ++s) {
        #pragma unroll
        for (int r = 0; r < 8; ++r) {
            __builtin_nontemporal_store(acc[s][r], Cp + (size_t)r * N + s * 16);
        }
    }
}

// ---------------------------------------------------------------------------
extern "C" void kernel_launch(void* const* d_in, const int* in_sizes, int n_in,
                              void* d_out, int out_size, void* d_ws, size_t ws_size,
                              hipStream_t stream) {
    const int*   x      = (const int*)d_in[0];
    const float* emb    = (const float*)d_in[1];
    const float* W_attn = (const float*)d_in[2];
    const float* U_attn = (const float*)d_in[3];
    const float* v_attn = (const float*)d_in[4];
    const float* W_ih   = (const float*)d_in[5];
    const float* b_ih   = (const float*)d_in[6];
    const float* W_hh   = (const float*)d_in[7];
    const float* b_hh   = (const float*)d_in[8];
    const float* W_out  = (const float*)d_in[9];
    const float* b_out  = (const float*)d_in[10];

    float* ws   = (float*)d_ws;
    float* hid  = ws;                               // B*T*H floats
    float* keys = ws + (size_t)Bb * Tt * Hh;        // B*T*H floats

    float* out     = (float*)d_out;                 // logits [B,T,V]
    float* h_final = out + (size_t)Bb * Tt * Vv;    // [B,H] tail

    rnn_attn_recurrent<<<Bb, 256, 0, stream>>>(
        x, emb, W_attn, U_attn, v_attn, W_ih, b_ih, W_hh, b_hh,
        hid, keys, h_final);

    logits_gemm_wmma<<<dim3(Vv / 256, (Bb * Tt) / 16), 128, 0, stream>>>(
        hid, W_out, b_out, out);
}